// StraightRenderer_68547678044273
// MI455X (gfx1250) — compile-verified
//
#include <hip/hip_runtime.h>
#include <hip/hip_bf16.h>

typedef __attribute__((ext_vector_type(2))) float v2f;
typedef __attribute__((ext_vector_type(8))) float v8f;

#define GSZ 1024
#define NSTROKE 16
#define NSEG 64
#define EPS_F 1e-5f

// xor-permute max within 16-lane rows via v_permlane16_b32 (VALU, not LDS pipe).
// Selector nibble i = i ^ mask, packed LSB-first; same pattern serves both rows.
__device__ __forceinline__ float rowxor_max(float v, unsigned s0, unsigned s1) {
    const unsigned u = __float_as_uint(v);
    const unsigned p = __builtin_amdgcn_permlane16(u, u, s0, s1, false, false);
    return fmaxf(v, __uint_as_float(p));
}

// One wave32 renders 64 consecutive pixels along h (4 tiles of 16, fixed g).
// WMMA f32 16x16x4 computes, per tile and segment-group j (k=j of each stroke):
//   dot[m][s] = px*ex + py*ey - (v.e)      (A row m = pixel: [px, py, 1, 0])
//   aff[m][s] = -2*px*vx - 2*py*vy + v2
// B operands are pre-swizzled into LDS per-lane so the hot loop is branchless.
// Per-stroke segment max stays in-lane; stroke max = 4 permlane16 xor-max steps.
__global__ __launch_bounds__(256) void StraightRenderer_kernel(
    const float* __restrict__ strokes,   // (16,5,2)
    const float* __restrict__ thick,     // (16,1)
    const float* __restrict__ colors,    // (16,3)
    float* __restrict__ out)             // (3,1024,1024)
{
    __shared__ v2f sBdot[4][32];        // [group j][lane] -> B operand (dot)
    __shared__ v2f sBaff[4][32];        // [group j][lane] -> B operand (aff)
    __shared__ v2f sDD[NSEG];           // (d2, 1/(d2+eps)) per segment
    __shared__ v2f sT[NSTROKE];         // (2t, 1/(2t)) per stroke
    __shared__ float sCol[3][NSTROKE];

    const int tid = threadIdx.x;

    // ---- per-block precompute: 256 threads -> 256 B-operand float2 entries ----
    {
        const int which = tid >> 7;        // 0 = dot, 1 = aff
        const int j     = (tid >> 5) & 3;  // segment group (k)
        const int lane0 = tid & 31;
        const int n0    = lane0 & 15;      // stroke column
        const int hi0   = lane0 >> 4;      // K-half
        const int s = n0, k = j;
        const float vx = fminf(fmaxf(strokes[(s * 5 + k) * 2 + 0], 0.f), 1.f) * (float)GSZ;
        const float vy = fminf(fmaxf(strokes[(s * 5 + k) * 2 + 1], 0.f), 1.f) * (float)GSZ;
        const float wx = fminf(fmaxf(strokes[(s * 5 + k + 1) * 2 + 0], 0.f), 1.f) * (float)GSZ;
        const float wy = fminf(fmaxf(strokes[(s * 5 + k + 1) * 2 + 1], 0.f), 1.f) * (float)GSZ;
        const float ex = wx - vx, ey = wy - vy;
        const float ve = vx * ex + vy * ey;
        const float v2 = vx * vx + vy * vy;
        v2f val;
        // dot B rows: [ex, ey, -ve, 0] ; aff B rows: [-2vx, -2vy, v2, 0]
        val.x = hi0 ? (which ? v2 : -ve) : (which ? -2.0f * vx : ex);
        val.y = hi0 ? 0.0f : (which ? -2.0f * vy : ey);
        if (which == 0) sBdot[j][lane0] = val;
        else            sBaff[j][lane0] = val;

        if (tid < NSEG) {   // per-segment scalars for segment 'tid' (recompute)
            const int ss = tid >> 2, kk = tid & 3;
            const float avx = fminf(fmaxf(strokes[(ss * 5 + kk) * 2 + 0], 0.f), 1.f) * (float)GSZ;
            const float avy = fminf(fmaxf(strokes[(ss * 5 + kk) * 2 + 1], 0.f), 1.f) * (float)GSZ;
            const float awx = fminf(fmaxf(strokes[(ss * 5 + kk + 1) * 2 + 0], 0.f), 1.f) * (float)GSZ;
            const float awy = fminf(fmaxf(strokes[(ss * 5 + kk + 1) * 2 + 1], 0.f), 1.f) * (float)GSZ;
            const float aex = awx - avx, aey = awy - avy;
            const float ad2 = aex * aex + aey * aey;
            v2f dd; dd.x = ad2; dd.y = 1.0f / (ad2 + EPS_F);
            sDD[tid] = dd;
        }
        if (tid < NSTROKE) {
            const float t = fmaxf(thick[tid] * 2.0f + 0.5f, 0.5f);
            v2f tt; tt.x = 2.0f * t; tt.y = 1.0f / (2.0f * t);
            sT[tid] = tt;
            sCol[0][tid] = fminf(fmaxf(colors[tid * 3 + 0], 0.f), 1.f);
            sCol[1][tid] = fminf(fmaxf(colors[tid * 3 + 1], 0.f), 1.f);
            sCol[2][tid] = fminf(fmaxf(colors[tid * 3 + 2], 0.f), 1.f);
        }
    }
    __syncthreads();

    // ---- tile assignment: wave -> (g, 64 pixels h0..h0+63) ----
    const int lane = tid & 31;
    const int wave = blockIdx.x * (blockDim.x >> 5) + (tid >> 5);
    const int g    = wave >> 4;            // 0..1023   (16 waves per row)
    const int h0   = (wave & 15) << 6;     // 0,64,...,960

    const int n  = lane & 15;  // D column: stroke id; also A row m = n
    const int hi = lane >> 4;  // 0 -> K0/K1, D rows m=0..7 ; 1 -> K2/K3, m=8..15

    // ---- hoisted LDS reads (branchless, one b64 load each) ----
    v2f bdot[4], baff[4], dd[4];
#pragma unroll
    for (int j = 0; j < 4; ++j) {
        bdot[j] = sBdot[j][lane];
        baff[j] = sBaff[j][lane];
        dd[j]   = sDD[n * 4 + j];          // (d2, inv_d2) of this lane's column
    }
    const v2f tT = sT[n];                  // (2t, 1/(2t))
    const float col0 = sCol[0][n], col1 = sCol[1][n], col2 = sCol[2][n];

    const float px  = (float)g;
    const float pxx = px * px;
    v8f zero = {0.f, 0.f, 0.f, 0.f, 0.f, 0.f, 0.f, 0.f};

#pragma unroll
    for (int t = 0; t < 4; ++t) {
        const int hbase = h0 + t * 16;

        // A operand: row m is pixel (g, hbase+m): [px, py, 1, 0]
        v2f a;
        a.x = hi ? 1.0f : px;
        a.y = hi ? 0.0f : (float)(hbase + n);

        // p2[r] = px^2 + py^2, float path only (no int mul/cvt chains)
        const float pyBase = (float)(hbase + hi * 8);
        float p2[8];
#pragma unroll
        for (int r = 0; r < 8; ++r) {
            const float py = pyBase + (float)r;
            p2[r] = fmaf(py, py, pxx);
        }

        float darkmax[8];
#pragma unroll
        for (int r = 0; r < 8; ++r) darkmax[r] = 0.0f;

#pragma unroll
        for (int j = 0; j < 4; ++j) {
            v8f dotD = __builtin_amdgcn_wmma_f32_16x16x4_f32(
                false, a, false, bdot[j], (short)0, zero, false, false);
            v8f affD = __builtin_amdgcn_wmma_f32_16x16x4_f32(
                false, a, false, baff[j], (short)0, zero, false, false);
            const float d2v   = dd[j].x;
            const float invd2 = dd[j].y;
#pragma unroll
            for (int r = 0; r < 8; ++r) {
                const float dot   = dotD[r];
                const float ttc   = fminf(fmaxf(dot * invd2, 0.0f), 1.0f);
                const float dist2 = p2[r] + affD[r] - 2.0f * ttc * dot + ttc * ttc * d2v;
                // raw v_sqrt_f32 (~1 ulp): feeds a clamped [0,1] ramp, precision ok
                const float dist  = __builtin_amdgcn_sqrtf(fmaxf(dist2, 0.0f));
                const float dark  = fminf(fmaxf((tT.x - dist) * tT.y, 0.0f), 1.0f);
                darkmax[r] = fmaxf(darkmax[r], dark);
            }
        }

        // ---- per-channel: scale by color, max over 16 strokes, store ----
        const size_t rowBase = (size_t)g * GSZ + (size_t)(hbase + hi * 8);
#pragma unroll
        for (int c = 0; c < 3; ++c) {
            const float col = (c == 0) ? col0 : ((c == 1) ? col1 : col2);
            float val[8];
#pragma unroll
            for (int r = 0; r < 8; ++r) val[r] = darkmax[r] * col;
            // xor-max butterfly over 16-lane rows, all on VALU (v_permlane16_b32)
#pragma unroll
            for (int r = 0; r < 8; ++r) {
                float v = val[r];
                v = rowxor_max(v, 0x67452301u, 0xEFCDAB89u);  // xor 1
                v = rowxor_max(v, 0x54761032u, 0xDCFE98BAu);  // xor 2
                v = rowxor_max(v, 0x32107654u, 0xBA98FEDCu);  // xor 4
                v = rowxor_max(v, 0xFEDCBA98u, 0x76543210u);  // xor 8
                val[r] = v;
            }
            // lane n (<8) of each half writes pixel m = hi*8 + n
            float outv = val[0];
#pragma unroll
            for (int r = 1; r < 8; ++r) outv = (n == r) ? val[r] : outv;
            if (n < 8)
                out[(size_t)c * GSZ * GSZ + rowBase + n] = outv;
        }
    }
}

extern "C" void kernel_launch(void* const* d_in, const int* in_sizes, int n_in,
                              void* d_out, int out_size, void* d_ws, size_t ws_size,
                              hipStream_t stream) {
    (void)in_sizes; (void)n_in; (void)d_ws; (void)ws_size; (void)out_size;
    const float* strokes = (const float*)d_in[0];  // (16,5,2)
    const float* thick   = (const float*)d_in[1];  // (16,1)
    const float* colors  = (const float*)d_in[2];  // (16,3)
    float* out = (float*)d_out;                    // (3,1024,1024)

    // 1M pixels / 64 px per wave = 16384 waves; 8 waves per 256-thread block.
    const int blocks = (GSZ * GSZ / 64) / 8;  // 2048
    StraightRenderer_kernel<<<blocks, 256, 0, stream>>>(strokes, thick, colors, out);
}